// FixedPointQuantizer_60266981097806
// MI455X (gfx1250) — compile-verified
//
#include <hip/hip_runtime.h>

// ---------------------------------------------------------------------------
// FixedPointQuantizer for MI455X (gfx1250)
//   pass 1: fused 8-candidate quantization-MSE reduction (async-to-LDS stream)
//   pass 2: argmin (tiny) then elementwise quantize with the winning f
// ---------------------------------------------------------------------------

typedef __attribute__((ext_vector_type(2))) float v2f;
typedef __attribute__((ext_vector_type(8))) float v8f;

#define THREADS 256
#define BLOCKS1 2048
#define BLOCKS2 8192

// ---- CDNA5 async copy: per-lane 16B global -> LDS, tracked by ASYNCcnt ----
__device__ __forceinline__ void async_load_f4(const float4* gptr, unsigned lds_off) {
    asm volatile("global_load_async_to_lds_b128 %0, %1, off"
                 :: "v"(lds_off), "v"((unsigned long long)gptr)
                 : "memory");
}
__device__ __forceinline__ void wait_async_le1() {
    asm volatile("s_wait_asynccnt 0x1" ::: "memory");
}
__device__ __forceinline__ void wait_async_le0() {
    asm volatile("s_wait_asynccnt 0x0" ::: "memory");
}

// scaled-space squared quantization error:  y = x * 2^f,
// clip bounds become constant [-128, 127], error measured in scaled units.
__device__ __forceinline__ void acc_err(float y, float& a) {
    float c = __builtin_amdgcn_fmed3f(y, -128.0f, 127.0f);
    float e = __builtin_rintf(c) - y;   // v_rndne_f32 (round half to even)
    a = fmaf(e, e, a);
}

__global__ __launch_bounds__(THREADS)
void fpq_mse_kernel(const float* __restrict__ x, long n, float* __restrict__ sums) {
    __shared__ float4 tile[2][THREADS];
    __shared__ float  lds_acc[8];

    const int tid = threadIdx.x;
    if (tid < 8) lds_acc[tid] = 0.0f;
    __syncthreads();

    float acc[8];
#pragma unroll
    for (int c = 0; c < 8; ++c) acc[c] = 0.0f;

    const long      n4      = n >> 2;
    const long      gstride = (long)gridDim.x * THREADS;
    const long      idx     = (long)blockIdx.x * THREADS + tid;
    const float4*   x4      = (const float4*)x;

    // raw LDS byte offsets of this thread's two slots (AS(3) ptrtoint)
    unsigned lds0 = (unsigned)(unsigned long long)
                    (__attribute__((address_space(3))) float4*)&tile[0][tid];
    unsigned lds1 = lds0 + THREADS * (unsigned)sizeof(float4);

    long iters = (idx < n4) ? ((n4 - 1 - idx) / gstride + 1) : 0;

    if (iters > 0) async_load_f4(&x4[idx], lds0);

    for (long it = 0; it < iters; ++it) {
        if (it + 1 < iters) {
            async_load_f4(&x4[idx + (it + 1) * gstride], (it & 1) ? lds0 : lds1);
            wait_async_le1();           // oldest (current) tile is resident
        } else {
            wait_async_le0();
        }
        float4 v = (it & 1) ? tile[1][tid] : tile[0][tid];
#pragma unroll
        for (int c = 0; c < 8; ++c) {
            float s = (float)(1u << c);          // 2^f
            acc_err(v.x * s, acc[c]);
            acc_err(v.y * s, acc[c]);
            acc_err(v.z * s, acc[c]);
            acc_err(v.w * s, acc[c]);
        }
    }

    // fold the 4^-f weighting (exact powers of two)
#pragma unroll
    for (int c = 0; c < 8; ++c) acc[c] *= 1.0f / (float)(1u << (2 * c));

    // ---- wave reduction ----
#if defined(__has_builtin) && __has_builtin(__builtin_amdgcn_wmma_f32_16x16x4_f32)
    // A = ones(16x4)  =>  D[m][n] = sum_k B[k][n] = b0[n] + b0[n+16] (+ b1 = 0)
    // i.e. one WMMA folds lanes n and n+16 of each accumulator.
    {
        v2f ones = {1.0f, 1.0f};
#pragma unroll
        for (int c = 0; c < 8; ++c) {
            v2f b = {acc[c], 0.0f};
            v8f cz = {};
            v8f d = __builtin_amdgcn_wmma_f32_16x16x4_f32(
                false, ones, false, b, (short)0, cz, false, false);
            acc[c] = d[0];   // lanes 16-31 now mirror lanes 0-15
        }
    }
#pragma unroll
    for (int c = 0; c < 8; ++c) {
        float s = acc[c];
        s += __shfl_xor(s, 1, 32);
        s += __shfl_xor(s, 2, 32);
        s += __shfl_xor(s, 4, 32);
        s += __shfl_xor(s, 8, 32);
        acc[c] = s;
    }
#else
#warning "wmma f32_16x16x4 builtin unavailable - falling back to full butterfly"
#pragma unroll
    for (int c = 0; c < 8; ++c) {
        float s = acc[c];
        s += __shfl_xor(s, 1, 32);
        s += __shfl_xor(s, 2, 32);
        s += __shfl_xor(s, 4, 32);
        s += __shfl_xor(s, 8, 32);
        s += __shfl_xor(s, 16, 32);
        acc[c] = s;
    }
#endif

    if ((tid & 31) == 0) {
#pragma unroll
        for (int c = 0; c < 8; ++c) atomicAdd(&lds_acc[c], acc[c]);  // ds_add_f32
    }
    __syncthreads();
    if (tid < 8) atomicAdd(&sums[tid], lds_acc[tid]);   // one per block per f
}

__global__ void fpq_argmin_kernel(const float* __restrict__ sums,
                                  float* __restrict__ params) {
    if (threadIdx.x == 0) {
        int   best = 0;
        float bv   = sums[0];
        for (int f = 1; f < 8; ++f) {
            float v = sums[f];
            if (v < bv) { bv = v; best = f; }   // strict < keeps first (jnp.argmin)
        }
        float step     = exp2f((float)(-best));
        float inv_step = exp2f((float)best);
        float r        = exp2f((float)(7 - best));
        params[0] = inv_step;
        params[1] = step;
        params[2] = -r;          // q_min
        params[3] = r - step;    // q_max
    }
}

__global__ __launch_bounds__(THREADS)
void fpq_quant_kernel(const float* __restrict__ x, float* __restrict__ out,
                      const float* __restrict__ params, long n) {
    const float inv_step = params[0];
    const float step     = params[1];
    const float qmin     = params[2];
    const float qmax     = params[3];

    const long    n4     = n >> 2;
    const long    stride = (long)gridDim.x * blockDim.x;
    const float4* x4     = (const float4*)x;
    float4*       o4     = (float4*)out;

    for (long i = (long)blockIdx.x * blockDim.x + threadIdx.x; i < n4; i += stride) {
        float4 v = x4[i];
        float4 q;
        q.x = __builtin_rintf(__builtin_amdgcn_fmed3f(v.x, qmin, qmax) * inv_step) * step;
        q.y = __builtin_rintf(__builtin_amdgcn_fmed3f(v.y, qmin, qmax) * inv_step) * step;
        q.z = __builtin_rintf(__builtin_amdgcn_fmed3f(v.z, qmin, qmax) * inv_step) * step;
        q.w = __builtin_rintf(__builtin_amdgcn_fmed3f(v.w, qmin, qmax) * inv_step) * step;
        o4[i] = q;
    }
}

extern "C" void kernel_launch(void* const* d_in, const int* in_sizes, int n_in,
                              void* d_out, int out_size, void* d_ws, size_t ws_size,
                              hipStream_t stream) {
    const float* x   = (const float*)d_in[0];
    float*       out = (float*)d_out;
    const long   n   = (long)in_sizes[0];

    float* sums   = (float*)d_ws;      // 8 candidate MSE sums
    float* params = sums + 8;          // inv_step, step, qmin, qmax

    hipMemsetAsync(d_ws, 0, 8 * sizeof(float), stream);
    fpq_mse_kernel<<<BLOCKS1, THREADS, 0, stream>>>(x, n, sums);
    fpq_argmin_kernel<<<1, 32, 0, stream>>>(sums, params);
    fpq_quant_kernel<<<BLOCKS2, THREADS, 0, stream>>>(x, out, params, n);
}